// LoadRouter_29308856828037
// MI455X (gfx1250) — compile-verified
//
#include <hip/hip_runtime.h>
#include <hip/hip_bf16.h>

#define NTOK   32768
#define HID    2880
#define NEXP   32

// K-block staged per step (floats per row)
#define KB     64
#define NKB    (HID / KB)            // 45
// LDS row stride in floats: bank = (4*row + col) & 63 -> conflict-free ds_load_b64
#define LSTR   68
// per-block tile: 4 waves * 16 tokens
#define ROWS   64
#define A_F    (ROWS * LSTR)         // 4352 floats
#define STAGEF (A_F + NEXP * LSTR)   // + 32*68 = 6528 floats = 26112 B
#define LDSBYTES (2 * STAGEF * 4)    // 52224 B double-buffered

typedef float v2f __attribute__((ext_vector_type(2)));
typedef float v8f __attribute__((ext_vector_type(8)));

// ---------------------------------------------------------------------------
// async global->LDS copy, 16B per lane, coalesced 512B per wave request.
// LDS byte offset comes from vdst, global address from the 64-bit vaddr.
// ---------------------------------------------------------------------------
__device__ __forceinline__ void async_copy_b128(unsigned lds_byte_off,
                                                const void* gptr) {
  asm volatile("global_load_async_to_lds_b128 %0, %1, off"
               :: "v"(lds_byte_off),
                  "v"((unsigned long long)(__SIZE_TYPE__)gptr)
               : "memory");
}

__device__ __forceinline__ void wait_async_le(int n) {
  if (n == 0) asm volatile("s_wait_asynccnt 0" ::: "memory");
  else        asm volatile("s_wait_asynccnt 12" ::: "memory");
}

// ---------------------------------------------------------------------------
// Phase 1: router GEMM, fp32 WMMA 16x16x4, async-LDS double-buffered pipeline.
// Block = 128 threads = 4 waves = 64 tokens; wave = 16 tokens x 32 experts.
// A frag (16x4 f32): lane<16 -> K={0,1}, lane>=16 -> K={2,3}, row M = lane&15.
// B frag (4x16 f32): same split, col N = lane&15.
// D (16x16 f32): VGPR j = row j / j+8 (half-wave), col lane&15.
// ---------------------------------------------------------------------------
__global__ void __launch_bounds__(128) router_gemm_wmma(
    const float* __restrict__ x, const float* __restrict__ w,
    float* __restrict__ logits) {
  extern __shared__ float sh[];

  const int tid       = threadIdx.x;
  const int lane      = tid & 31;
  const int waveInBlk = tid >> 5;
  const int blockRow0 = blockIdx.x * ROWS;
  const int rowBase   = blockRow0 + waveInBlk * 16;
  const int m         = lane & 15;
  const int koff      = (lane >> 4) << 1;  // 0 or 2

  // ---- stage(kb, buf): 12 async b128 ops per thread --------------------
  auto stage = [&](int kb, int buf) {
    const unsigned bufF = (unsigned)buf * STAGEF;
    // A tile: 64 rows x 256B, 16 lanes per row (coalesced)
#pragma unroll
    for (int it = 0; it < 8; ++it) {
      unsigned g   = (unsigned)(it * 128 + tid);  // 0..1023
      unsigned row = g >> 4;                      // 0..63
      unsigned c16 = g & 15u;
      const float* gp = x + (__SIZE_TYPE__)(blockRow0 + row) * HID + kb * KB + c16 * 4;
      async_copy_b128((bufF + row * LSTR + c16 * 4u) * 4u, gp);
    }
    // W tile: 32 rows x 256B
#pragma unroll
    for (int it = 0; it < 4; ++it) {
      unsigned g   = (unsigned)(it * 128 + tid);  // 0..511
      unsigned row = g >> 4;                      // 0..31
      unsigned c16 = g & 15u;
      const float* gp = w + (__SIZE_TYPE__)row * HID + kb * KB + c16 * 4;
      async_copy_b128((bufF + A_F + row * LSTR + c16 * 4u) * 4u, gp);
    }
  };

  v8f acc0 = {0.f,0.f,0.f,0.f,0.f,0.f,0.f,0.f};
  v8f acc1 = {0.f,0.f,0.f,0.f,0.f,0.f,0.f,0.f};

  stage(0, 0);  // prologue DMA

  for (int kb = 0; kb < NKB; ++kb) {
    const int prefetch = (kb + 1 < NKB);
    if (prefetch) stage(kb + 1, (kb + 1) & 1);   // overlap next DMA with compute
    wait_async_le(prefetch ? 12 : 0);            // current buffer's 12 ops done
    __syncthreads();                             // visible to all 4 waves

    const int bufF = (kb & 1) * STAGEF;
    const float* As  = sh + bufF + (waveInBlk * 16 + m) * LSTR;
    const float* Ws0 = sh + bufF + A_F + m * LSTR;
    const float* Ws1 = sh + bufF + A_F + (m + 16) * LSTR;
#pragma unroll
    for (int kk = 0; kk < KB; kk += 4) {
      v2f a  = *(const v2f*)(As  + kk + koff);
      v2f b0 = *(const v2f*)(Ws0 + kk + koff);
      v2f b1 = *(const v2f*)(Ws1 + kk + koff);
      acc0 = __builtin_amdgcn_wmma_f32_16x16x4_f32(false, a, false, b0,
                                                   (short)0, acc0, false, false);
      acc1 = __builtin_amdgcn_wmma_f32_16x16x4_f32(false, a, false, b1,
                                                   (short)0, acc1, false, false);
    }
    __syncthreads();  // all waves done reading buf before it is re-filled
  }

  const int n    = lane & 15;
  const int rAdd = (lane >> 4) << 3;  // 0 or 8
#pragma unroll
  for (int j = 0; j < 8; ++j) {
    const __SIZE_TYPE__ r = (__SIZE_TYPE__)(rowBase + rAdd + j) * NEXP;
    logits[r + n]      = acc0[j];
    logits[r + 16 + n] = acc1[j];
  }
}

// ---------------------------------------------------------------------------
// JAX threefry2x32, key = (0, 42)  (jax.random.key(42))
// ---------------------------------------------------------------------------
__device__ __forceinline__ unsigned rotl32(unsigned v, unsigned d) {
  return (v << d) | (v >> (32u - d));
}

__device__ __forceinline__ void threefry2x32_k42(unsigned c0, unsigned c1,
                                                 unsigned& o0, unsigned& o1) {
  const unsigned k0 = 0u, k1 = 42u;
  const unsigned k2 = k0 ^ k1 ^ 0x1BD11BDAu;
  unsigned x0 = c0 + k0;
  unsigned x1 = c1 + k1;
#define TF_QR4(a, b, c, d)                         \
  { x0 += x1; x1 = rotl32(x1, a); x1 ^= x0;        \
    x0 += x1; x1 = rotl32(x1, b); x1 ^= x0;        \
    x0 += x1; x1 = rotl32(x1, c); x1 ^= x0;        \
    x0 += x1; x1 = rotl32(x1, d); x1 ^= x0; }
  TF_QR4(13u, 15u, 26u, 6u);  x0 += k1; x1 += k2 + 1u;
  TF_QR4(17u, 29u, 16u, 24u); x0 += k2; x1 += k0 + 2u;
  TF_QR4(13u, 15u, 26u, 6u);  x0 += k0; x1 += k1 + 3u;
  TF_QR4(17u, 29u, 16u, 24u); x0 += k1; x1 += k2 + 4u;
  TF_QR4(13u, 15u, 26u, 6u);  x0 += k2; x1 += k0 + 5u;
#undef TF_QR4
  o0 = x0; o1 = x1;
}

// ---------------------------------------------------------------------------
// Phase 2: bias add + top-4 + threefry subsample + sort + softmax + scatter.
// ---------------------------------------------------------------------------
__global__ void __launch_bounds__(256) router_topk(
    const float* __restrict__ logits, const float* __restrict__ bias,
    float* __restrict__ scores, int* __restrict__ out_idx) {
  const int t = blockIdx.x * blockDim.x + threadIdx.x;
  if (t >= NTOK) return;

  const float* row = logits + (__SIZE_TYPE__)t * NEXP;

  // top-4, descending, stable (strict '>' -> lower index wins ties, = lax.top_k)
  float tv0 = -__builtin_inff(), tv1 = tv0, tv2 = tv0, tv3 = tv0;
  int   ti0 = 0, ti1 = 0, ti2 = 0, ti3 = 0;
#pragma unroll
  for (int e = 0; e < NEXP; ++e) {
    float v = row[e] + bias[e];
    if (v > tv3) {
      if (v > tv0)      { tv3=tv2; ti3=ti2; tv2=tv1; ti2=ti1; tv1=tv0; ti1=ti0; tv0=v; ti0=e; }
      else if (v > tv1) { tv3=tv2; ti3=ti2; tv2=tv1; ti2=ti1; tv1=v; ti1=e; }
      else if (v > tv2) { tv3=tv2; ti3=ti2; tv2=v; ti2=e; }
      else              { tv3=v; ti3=e; }
    }
  }
  float tvals[4] = {tv0, tv1, tv2, tv3};
  int   tinds[4] = {ti0, ti1, ti2, ti3};

  // noise = jax.random.uniform(key(42),(N,4)): counts=iota(N*4) split in halves
  const unsigned H = (unsigned)(NTOK * 4 / 2);
  float nz[4];
#pragma unroll
  for (int j = 0; j < 4; ++j) {
    unsigned f = (unsigned)t * 4u + (unsigned)j;
    unsigned o0, o1;
    if (f < H) { threefry2x32_k42(f, f + H, o0, o1); }
    else       { threefry2x32_k42(f - H, f, o0, o1); o0 = o1; }
    nz[j] = __uint_as_float((o0 >> 9) | 0x3f800000u) - 1.0f;
  }

  // stable ascending argsort of 4; keep first two
  int ord[4] = {0, 1, 2, 3};
#pragma unroll
  for (int i = 1; i < 4; ++i) {
    int oi = ord[i];
    float vi = nz[oi];
    int j = i - 1;
    while (j >= 0 && nz[ord[j]] > vi) { ord[j + 1] = ord[j]; --j; }
    ord[j + 1] = oi;
  }
  float sv0 = tvals[ord[0]], sv1 = tvals[ord[1]];
  int   si0 = tinds[ord[0]], si1 = tinds[ord[1]];

  if (sv1 > sv0) {  // argsort(-v) stable: swap only if strictly greater
    float tf = sv0; sv0 = sv1; sv1 = tf;
    int   ii = si0; si0 = si1; si1 = ii;
  }

  float e0 = expf(sv0 - sv0);
  float e1 = expf(sv1 - sv0);
  float inv = 1.0f / (e0 + e1);
  float pr0 = e0 * inv, pr1 = e1 * inv;

  float* orow = scores + (__SIZE_TYPE__)t * NEXP;
#pragma unroll
  for (int e = 0; e < NEXP; ++e) orow[e] = 0.0f;
  orow[si0] = pr0;
  orow[si1] = pr1;
  out_idx[2 * t]     = si0;
  out_idx[2 * t + 1] = si1;
}

extern "C" void kernel_launch(void* const* d_in, const int* in_sizes, int n_in,
                              void* d_out, int out_size, void* d_ws, size_t ws_size,
                              hipStream_t stream) {
  const float* x = (const float*)d_in[0];   // [32768, 2880] f32
  const float* w = (const float*)d_in[1];   // [32, 2880]    f32
  const float* b = (const float*)d_in[2];   // [32]          f32

  float* scores = (float*)d_out;                               // [32768, 32]
  int*   oidx   = (int*)((float*)d_out + (__SIZE_TYPE__)NTOK * NEXP); // [32768, 2]
  float* logits = (float*)d_ws;                                // 4 MB scratch

  router_gemm_wmma<<<NTOK / ROWS, 128, LDSBYTES, stream>>>(x, w, logits);
  router_topk<<<NTOK / 256, 256, 0, stream>>>(logits, b, scores, oidx);
}